// NGram_Attention_15702400434701
// MI455X (gfx1250) — compile-verified
//
#include <hip/hip_runtime.h>
#include <hip/hip_bf16.h>
#include <math.h>

// Problem constants (from reference): B=16, T=4096, D=512, N=3 levels.
#define Bsz  16
#define Tlen 4096
#define Dm   512
#define NLev 3

typedef float v2f __attribute__((ext_vector_type(2)));
typedef float v8f __attribute__((ext_vector_type(8)));

// ---------------------------------------------------------------------------
// K1: q_i[b,d] = sum_e s_prev[b,e] * Vw[i][e,d]   ([16x512] @ [512x512], x3)
// One wave per (level, 16-wide n tile). V_WMMA_F32_16X16X4_F32, K-loop of 128.
// A-frag lane(h,lo): vg -> A[M=lo, K=k0+2h+vg]; B-frag: vg -> B[K=k0+2h+vg, N=n0+lo]
// C-frag: c[r] = C[M=r+8h, N=n0+lo]
// ---------------------------------------------------------------------------
__global__ __launch_bounds__(32) void k_qproj(const float* __restrict__ s_prev,
                                              const float* __restrict__ Vw,
                                              float* __restrict__ qbuf) {
  const int lane = threadIdx.x;
  const int h = lane >> 4, lo = lane & 15;
  const int n0 = blockIdx.x * 16;
  const int i  = blockIdx.y;
  const float* W = Vw + (size_t)i * Dm * Dm;
  v8f c = {0.f,0.f,0.f,0.f,0.f,0.f,0.f,0.f};
#pragma unroll 4
  for (int k0 = 0; k0 < Dm; k0 += 4) {
    v2f a, b;
    const float* ap = s_prev + (size_t)lo * Dm + k0 + 2 * h;
    a[0] = ap[0]; a[1] = ap[1];
    const float* bp = W + (size_t)(k0 + 2 * h) * Dm + n0 + lo;
    b[0] = bp[0]; b[1] = bp[Dm];
    c = __builtin_amdgcn_wmma_f32_16x16x4_f32(false, a, false, b, (short)0, c,
                                              false, false);
  }
  float* qo = qbuf + (size_t)i * Bsz * Dm;
#pragma unroll
  for (int r = 0; r < 8; ++r)
    qo[(size_t)(r + 8 * h) * Dm + n0 + lo] = c[r];
}

// ---------------------------------------------------------------------------
// K1b: c_i[b] = s_prev[b] . Vb[i]   (one wave per (b,i))
// ---------------------------------------------------------------------------
__global__ __launch_bounds__(32) void k_cbias(const float* __restrict__ s_prev,
                                              const float* __restrict__ Vb,
                                              float* __restrict__ cb) {
  const int b = blockIdx.x, i = blockIdx.y;
  const int lane = threadIdx.x;
  float s = 0.f;
#pragma unroll
  for (int e = lane; e < Dm; e += 32) s += s_prev[b * Dm + e] * Vb[i * Dm + e];
#pragma unroll
  for (int off = 16; off > 0; off >>= 1) s += __shfl_xor(s, off, 32);
  if (lane == 0) cb[i * Bsz + b] = s;
}

// ---------------------------------------------------------------------------
// K2: raw_i[b,t] = enc_hs[b,t] . q_i[b]  for all 3 levels, one enc_hs pass.
// One wave per 8 rows; lane owns a 16-float slice of D; float4 streaming loads.
// ---------------------------------------------------------------------------
__global__ __launch_bounds__(256) void k_dots(const float* __restrict__ enc,
                                              const float* __restrict__ qbuf,
                                              float* __restrict__ raw) {
  const int b = blockIdx.x;
  const int lane = threadIdx.x & 31;
  const int wave = threadIdx.x >> 5;
  const int t0 = blockIdx.y * 64 + wave * 8;

  float ql[NLev][16];
#pragma unroll
  for (int i = 0; i < NLev; ++i) {
    const float4* qp =
        (const float4*)(qbuf + ((size_t)i * Bsz + b) * Dm + lane * 16);
#pragma unroll
    for (int c4 = 0; c4 < 4; ++c4) {
      float4 v = qp[c4];
      ql[i][4 * c4 + 0] = v.x; ql[i][4 * c4 + 1] = v.y;
      ql[i][4 * c4 + 2] = v.z; ql[i][4 * c4 + 3] = v.w;
    }
  }
  for (int r = 0; r < 8; ++r) {
    const int t = t0 + r;
    const float4* ep =
        (const float4*)(enc + ((size_t)b * Tlen + t) * Dm + lane * 16);
    float a0 = 0.f, a1 = 0.f, a2 = 0.f;
#pragma unroll
    for (int c4 = 0; c4 < 4; ++c4) {
      float4 e = ep[c4];
      a0 += e.x * ql[0][4*c4] + e.y * ql[0][4*c4+1] + e.z * ql[0][4*c4+2] + e.w * ql[0][4*c4+3];
      a1 += e.x * ql[1][4*c4] + e.y * ql[1][4*c4+1] + e.z * ql[1][4*c4+2] + e.w * ql[1][4*c4+3];
      a2 += e.x * ql[2][4*c4] + e.y * ql[2][4*c4+1] + e.z * ql[2][4*c4+2] + e.w * ql[2][4*c4+3];
    }
#pragma unroll
    for (int off = 16; off > 0; off >>= 1) {
      a0 += __shfl_xor(a0, off, 32);
      a1 += __shfl_xor(a1, off, 32);
      a2 += __shfl_xor(a2, off, 32);
    }
    if (lane == 0) {
      raw[((size_t)0 * Bsz + b) * Tlen + t] = a0;
      raw[((size_t)1 * Bsz + b) * Tlen + t] = a1;
      raw[((size_t)2 * Bsz + b) * Tlen + t] = a2;
    }
  }
}

// ---------------------------------------------------------------------------
// K3: per (b, level): logits = window-mean(raw) + cbias, masked softmax,
// then reversed-window weights w_i[b,t] = (1/k) * sum att[t-k+1..t].
// Level 0 attention also goes straight to d_out.
// ---------------------------------------------------------------------------
__global__ __launch_bounds__(256) void k_softmax_win(
    const float* __restrict__ raw, const float* __restrict__ cb,
    const unsigned char* __restrict__ mask, float* __restrict__ attb,
    float* __restrict__ wbuf, float* __restrict__ out_att) {
  const int b = blockIdx.x;
  const int i = blockIdx.y;
  const int L = Tlen - i;
  const int tid = threadIdx.x;
  const float invk = 1.0f / (float)(i + 1);
  const float cbias = cb[i * Bsz + b];
  const float* rw = raw + ((size_t)i * Bsz + b) * Tlen;
  const unsigned char* mk = mask + (size_t)b * Tlen;
  float* ab = attb + ((size_t)i * Bsz + b) * Tlen;
  __shared__ float red[256];
  const float NEG_INF = -__builtin_inff();

  float lmax = NEG_INF;
  for (int l = tid; l < L; l += 256) {
    float s = 0.f; bool m = false;
    for (int j = 0; j <= i; ++j) { s += rw[l + j]; m = m || (mk[l + j] != 0); }
    float lg = m ? NEG_INF : (s * invk + cbias);
    ab[l] = lg;
    lmax = fmaxf(lmax, lg);
  }
  red[tid] = lmax; __syncthreads();
  for (int s = 128; s > 0; s >>= 1) {
    if (tid < s) red[tid] = fmaxf(red[tid], red[tid + s]);
    __syncthreads();
  }
  const float gmax = red[0]; __syncthreads();

  float lsum = 0.f;
  for (int l = tid; l < L; l += 256) {
    float e = expf(ab[l] - gmax);  // exp(-inf)=0 handles masked slots
    ab[l] = e; lsum += e;
  }
  red[tid] = lsum; __syncthreads();
  for (int s = 128; s > 0; s >>= 1) {
    if (tid < s) red[tid] += red[tid + s];
    __syncthreads();
  }
  const float inv = 1.0f / red[0];

  for (int l = tid; l < L; l += 256) {
    float a = ab[l] * inv;
    ab[l] = a;
    if (i == 0) out_att[(size_t)b * Tlen + l] = a;
  }
  __syncthreads();

  float* wb = wbuf + ((size_t)i * Bsz + b) * Tlen;
  for (int t = tid; t < Tlen; t += 256) {
    float acc = 0.f;
    for (int j = 0; j <= i; ++j) {
      int l = t - j;
      if (l >= 0 && l < L) acc += ab[l];
    }
    wb[t] = acc * invk;
  }
}

// ---------------------------------------------------------------------------
// K4: pooled partials (t-chunked, fixed order -> deterministic, no atomics).
// ---------------------------------------------------------------------------
__global__ __launch_bounds__(256) void k_pool_partial(
    const float* __restrict__ enc, const float* __restrict__ wbuf,
    float* __restrict__ partial) {
  const int b = blockIdx.x;
  const int d = blockIdx.y * 256 + threadIdx.x;
  const int tc = blockIdx.z;
  const int tbeg = tc * (Tlen / 32);
  const float* w0 = wbuf + ((size_t)0 * Bsz + b) * Tlen;
  const float* w1 = wbuf + ((size_t)1 * Bsz + b) * Tlen;
  const float* w2 = wbuf + ((size_t)2 * Bsz + b) * Tlen;
  float a0 = 0.f, a1 = 0.f, a2 = 0.f;
  for (int t = tbeg; t < tbeg + Tlen / 32; ++t) {
    float e = enc[((size_t)b * Tlen + t) * Dm + d];
    a0 += w0[t] * e; a1 += w1[t] * e; a2 += w2[t] * e;
  }
  float* p = partial + (size_t)tc * (NLev * Bsz * Dm);
  p[((size_t)0 * Bsz + b) * Dm + d] = a0;
  p[((size_t)1 * Bsz + b) * Dm + d] = a1;
  p[((size_t)2 * Bsz + b) * Dm + d] = a2;
}

__global__ __launch_bounds__(256) void k_pool_reduce(
    const float* __restrict__ partial, float* __restrict__ pooled) {
  const int idx = blockIdx.x * 256 + threadIdx.x;  // < 3*16*512
  float s = 0.f;
#pragma unroll
  for (int tc = 0; tc < 32; ++tc)
    s += partial[(size_t)tc * (NLev * Bsz * Dm) + idx];
  pooled[idx] = s;
}

// ---------------------------------------------------------------------------
// K5: ctx[b,d] = sum_i ( sum_e pooled_i[b,e] * Ww[i][d,e] + Wb[i][d] ).
// WMMA: A = pooled_i tile (M=batch, K=e), B[k][n] = Ww[i][n][e0+k].
// ---------------------------------------------------------------------------
__global__ __launch_bounds__(32) void k_ctx(const float* __restrict__ pooled,
                                            const float* __restrict__ Ww,
                                            const float* __restrict__ Wb,
                                            float* __restrict__ out_ctx) {
  const int lane = threadIdx.x;
  const int h = lane >> 4, lo = lane & 15;
  const int n0 = blockIdx.x * 16;
  v8f c = {0.f,0.f,0.f,0.f,0.f,0.f,0.f,0.f};
  for (int i = 0; i < NLev; ++i) {
    const float* P = pooled + (size_t)i * Bsz * Dm;
    const float* W = Ww + (size_t)i * Dm * Dm;
#pragma unroll 4
    for (int e0 = 0; e0 < Dm; e0 += 4) {
      v2f a, bfr;
      const float* ap = P + (size_t)lo * Dm + e0 + 2 * h;
      a[0] = ap[0]; a[1] = ap[1];
      const float* bp = W + (size_t)(n0 + lo) * Dm + e0 + 2 * h;
      bfr[0] = bp[0]; bfr[1] = bp[1];
      c = __builtin_amdgcn_wmma_f32_16x16x4_f32(false, a, false, bfr, (short)0,
                                                c, false, false);
    }
  }
  const int n = n0 + lo;
  const float bias = Wb[n] + Wb[Dm + n] + Wb[2 * Dm + n];
#pragma unroll
  for (int r = 0; r < 8; ++r)
    out_ctx[(size_t)(r + 8 * h) * Dm + n] = c[r] + bias;
}

// ---------------------------------------------------------------------------
extern "C" void kernel_launch(void* const* d_in, const int* in_sizes, int n_in,
                              void* d_out, int out_size, void* d_ws,
                              size_t ws_size, hipStream_t stream) {
  (void)in_sizes; (void)n_in; (void)out_size; (void)ws_size;
  const float* s_prev = (const float*)d_in[0];                 // [B,D]
  const float* enc_hs = (const float*)d_in[1];                 // [B,T,D]
  const unsigned char* src_mask = (const unsigned char*)d_in[2]; // [B,T] bool
  const float* Vw = (const float*)d_in[3];                     // [N,D,D]
  const float* Vb = (const float*)d_in[4];                     // [N,D]
  const float* Ww = (const float*)d_in[5];                     // [N,D,D]
  const float* Wb = (const float*)d_in[6];                     // [N,D]

  float* out_ctx = (float*)d_out;                // [B,D] = 8192 floats
  float* out_att = (float*)d_out + Bsz * Dm;     // [B,T] = 65536 floats

  // Workspace carve-up (floats)
  float* ws = (float*)d_ws;
  float* qbuf    = ws;                       // 3*16*512      = 24576
  float* cb      = qbuf + NLev * Bsz * Dm;   // 48 (+pad)
  float* raw     = cb + 64;                  // 3*16*4096     = 196608
  float* attb    = raw + NLev * Bsz * Tlen;  // 196608
  float* wbuf    = attb + NLev * Bsz * Tlen; // 196608
  float* partial = wbuf + NLev * Bsz * Tlen; // 32*3*16*512   = 786432
  float* pooled  = partial + 32 * NLev * Bsz * Dm; // 24576
  // total ~1.43M floats ~ 5.7 MB

  k_qproj<<<dim3(Dm / 16, NLev), 32, 0, stream>>>(s_prev, Vw, qbuf);
  k_cbias<<<dim3(Bsz, NLev), 32, 0, stream>>>(s_prev, Vb, cb);
  k_dots<<<dim3(Bsz, Tlen / 64), 256, 0, stream>>>(enc_hs, qbuf, raw);
  k_softmax_win<<<dim3(Bsz, NLev), 256, 0, stream>>>(raw, cb, src_mask, attb,
                                                     wbuf, out_att);
  k_pool_partial<<<dim3(Bsz, Dm / 256, 32), 256, 0, stream>>>(enc_hs, wbuf,
                                                              partial);
  k_pool_reduce<<<(NLev * Bsz * Dm) / 256, 256, 0, stream>>>(partial, pooled);
  k_ctx<<<Dm / 16, 32, 0, stream>>>(pooled, Ww, Wb, out_ctx);
}